// FeatureRadiusDGCNN_52553219833919
// MI455X (gfx1250) — compile-verified
//
#include <hip/hip_runtime.h>

#define BB   8
#define NN   2048
#define KNBR 20
#define BN   (BB*NN)
#define KC   32

typedef __attribute__((ext_vector_type(2))) float v2f;
typedef __attribute__((ext_vector_type(8))) float v8f;

__device__ __forceinline__ float lrelu(float v){ return v >= 0.f ? v : 0.2f*v; }

// monotone encode of float into unsigned for atomicMax
__device__ __forceinline__ unsigned fenc(float f){
  unsigned u = __float_as_uint(f);
  return u ^ ((unsigned)((int)u >> 31) | 0x80000000u);
}
__device__ __forceinline__ float fdec(unsigned k){
  unsigned u = (k & 0x80000000u) ? (k ^ 0x80000000u) : ~k;
  return __uint_as_float(u);
}
#define ENC_NEGINF 0x007FFFFFu   // fenc(-inf)

__device__ __forceinline__ v8f wmma4(v2f a, v2f b, v8f c){
  // D = A(16x4,f32) * B(4x16,f32) + C(16x16,f32)
  return __builtin_amdgcn_wmma_f32_16x16x4_f32(false, a, false, b, (short)0, c, false, false);
}

// ---------------------------------------------------------------- utilities

__global__ void zero_f(float* p, int n){
  int i = blockIdx.x*blockDim.x + threadIdx.x;
  if (i < n) p[i] = 0.f;
}

// x: (B,3,N) channel-major -> xt0: [b][n][3]
__global__ void transpose_x(const float* __restrict__ x, float* __restrict__ xt0){
  int e = blockIdx.x*blockDim.x + threadIdx.x;
  if (e >= BN*3) return;
  int pt = e / 3, c = e % 3;
  int b = pt / NN, n = pt % NN;
  xt0[e] = x[((size_t)b*3 + c)*NN + n];
}

// per-point validity count; counts[0] = sum(valid) (edge BN n), counts[1] = sum(ok) (point BN n)
__global__ void ok_kernel(const unsigned char* __restrict__ valid, int* __restrict__ okBuf,
                          float* __restrict__ counts){
  int pt = blockIdx.x*blockDim.x + threadIdx.x;
  if (pt >= BN) return;
  int c = 0;
  for (int k = 0; k < KNBR; ++k) c += valid[(size_t)pt*KNBR + k] ? 1 : 0;
  okBuf[pt] = (c >= 1) ? 1 : 0;
  atomicAdd(&counts[0], (float)c);
  atomicAdd(&counts[1], (c >= 1) ? 1.f : 0.f);
}

__global__ void finalize_bn(const float* __restrict__ sum, const float* __restrict__ sq,
                            const float* __restrict__ counts, int cidx,
                            const float* __restrict__ gamma, const float* __restrict__ beta,
                            float* __restrict__ scale, float* __restrict__ shift, int O){
  int c = blockIdx.x*blockDim.x + threadIdx.x;
  if (c >= O) return;
  float n    = fmaxf(counts[cidx], 1.f);
  float mean = sum[c] / n;
  float var  = fmaxf(sq[c] / n - mean*mean, 0.f);
  float inv  = rsqrtf(var + 1e-5f);
  float s    = gamma[c] * inv;
  scale[c] = s;
  shift[c] = beta[c] - mean * s;
}

// elementwise BN-apply + lrelu on materialized h: [M][O]
__global__ void bn_act(const float* __restrict__ h, const float* __restrict__ scale,
                       const float* __restrict__ shift, float* __restrict__ out, int total, int O){
  int e = blockIdx.x*blockDim.x + threadIdx.x;
  if (e >= total) return;
  int c = e % O;
  out[e] = lrelu(h[e]*scale[c] + shift[c]);
}

__global__ void concat_kernel(const float* __restrict__ a, const float* __restrict__ b,
                              const float* __restrict__ c, const float* __restrict__ d,
                              float* __restrict__ out){
  size_t e = (size_t)blockIdx.x*blockDim.x + threadIdx.x;
  if (e >= (size_t)BN*512) return;
  size_t pt = e >> 9; int ch = (int)(e & 511);
  float v;
  if      (ch <  64) v = a[pt*64  + ch];
  else if (ch < 128) v = b[pt*64  + ch - 64];
  else if (ch < 256) v = c[pt*128 + ch - 128];
  else               v = d[pt*256 + ch - 256];
  out[e] = v;
}

// h: [pt][32] -> d_out: (B,32,N)
__global__ void write_out(const float* __restrict__ h, float* __restrict__ out){
  int e = blockIdx.x*blockDim.x + threadIdx.x;
  if (e >= BN*32) return;
  int pt = e >> 5, o = e & 31;
  int b = pt / NN, n = pt % NN;
  out[((size_t)b*32 + o)*NN + n] = h[e];
}

// ------------------------------------------- edge block 1 (Cin=3, scalar path)

template<int PHASE>
__global__ __launch_bounds__(256)
void edge1(const float* __restrict__ xt0, const int* __restrict__ idx,
           const unsigned char* __restrict__ valid, const float* __restrict__ W, // [64][6]
           float* __restrict__ statSum, float* __restrict__ statSq,
           const float* __restrict__ scale, const float* __restrict__ shift,
           const int* __restrict__ okBuf, float* __restrict__ out /*[BN][64]*/){
  __shared__ float sF[4][KNBR][6];
  __shared__ float sM[4][KNBR];
  __shared__ float sW[64*6];
  __shared__ float sSum[64], sSq[64];
  int tid  = threadIdx.x;
  int base = blockIdx.x * 4;
  for (int e = tid; e < 384; e += 256) sW[e] = W[e];
  if (PHASE == 0 && tid < 64){ sSum[tid] = 0.f; sSq[tid] = 0.f; }
  if (tid < 4*KNBR){
    int p = tid / KNBR, k = tid % KNBR;
    int pt = base + p;
    int b = pt / NN;
    int j = idx[(size_t)pt*KNBR + k];
    float m = valid[(size_t)pt*KNBR + k] ? 1.f : 0.f;
    const float* ctr = xt0 + (size_t)pt*3;
    const float* nbr = xt0 + ((size_t)b*NN + j)*3;
    for (int c = 0; c < 3; ++c){
      sF[p][k][c]   = (nbr[c] - ctr[c]) * m;
      sF[p][k][c+3] = ctr[c] * m;
    }
    sM[p][k] = m;
  }
  __syncthreads();
  int p = tid >> 6, o = tid & 63;
  if (PHASE == 0){
    float s1 = 0.f, s2 = 0.f;
    for (int k = 0; k < KNBR; ++k){
      float h = 0.f;
      #pragma unroll
      for (int c = 0; c < 6; ++c) h += sW[o*6+c] * sF[p][k][c];
      s1 += h; s2 += h*h;
    }
    atomicAdd(&sSum[o], s1); atomicAdd(&sSq[o], s2);
    __syncthreads();
    if (tid < 64){ atomicAdd(&statSum[tid], sSum[tid]); atomicAdd(&statSq[tid], sSq[tid]); }
  } else {
    float sc = scale[o], sh = shift[o];
    float mx = -__builtin_inff();
    for (int k = 0; k < KNBR; ++k){
      float h = 0.f;
      #pragma unroll
      for (int c = 0; c < 6; ++c) h += sW[o*6+c] * sF[p][k][c];
      h = lrelu(h*sc + sh);
      if (sM[p][k] > 0.f) mx = fmaxf(mx, h);
    }
    int pt = base + p;
    out[(size_t)pt*64 + o] = okBuf[pt] ? mx : 0.f;
  }
}

// ----------------------------- edge blocks 2-4: fused gather + WMMA f32 GEMM
// M-tile = 128 edge slots = 4 points x 32 k-slots (20 real, 12 zero-masked)
// N-tile = 64 out channels (4x 16x16 WMMA tiles per wave), K-chunks of 32.
// W tile stored K-pair interleaved so each B fragment is one ds_load_b64.
// PHASE 0: accumulate per-channel sum/sumsq.  PHASE 1: BN+lrelu+masked-max -> features.

template<int PHASE>
__global__ __launch_bounds__(256)
void edge_gemm(const float* __restrict__ xin, const int* __restrict__ idx,
               const unsigned char* __restrict__ valid, const float* __restrict__ Wt, // [Cout][2*Cin]
               float* __restrict__ statSum, float* __restrict__ statSq,
               const float* __restrict__ scale, const float* __restrict__ shift,
               const int* __restrict__ okBuf, float* __restrict__ feat_out, // [BN][Cout]
               int Cin, int Cout){
  const int Cin2 = 2*Cin;
  __shared__ float sF[128][KC+4];
  __shared__ v2f   sW2[KC/2][64];        // sW2[k/2][o] = {W[k][o], W[k+1][o]}
  __shared__ int   sIdx[4][32];
  __shared__ float sMask[128];
  __shared__ float sSum[64], sSq[64];
  __shared__ unsigned sMaxKey[4][64];

  int tid  = threadIdx.x;
  int lane = tid & 31;
  int w    = tid >> 5;
  int pg   = blockIdx.x;
  int b    = pg / (NN/4);
  int n0   = (pg % (NN/4)) * 4;
  int O0   = blockIdx.y * 64;
  size_t ptbase = (size_t)b*NN + n0;

  if (tid < 128){
    int p = tid >> 5, ks = tid & 31;
    size_t pt = ptbase + p;
    int j = 0; float m = 0.f;
    if (ks < KNBR){
      j = idx[pt*KNBR + ks];
      m = valid[pt*KNBR + ks] ? 1.f : 0.f;
    }
    sIdx[p][ks] = j;
    sMask[tid]  = m;
  }
  if (PHASE == 0){ if (tid < 64){ sSum[tid] = 0.f; sSq[tid] = 0.f; } }
  else           { ((unsigned*)sMaxKey)[tid] = ENC_NEGINF; }
  __syncthreads();

  v8f acc[4] = {};
  for (int c0 = 0; c0 < Cin2; c0 += KC){
    __syncthreads();
    #pragma unroll
    for (int r = 0; r < 4; ++r){                 // B tile: 1024 K-pairs
      int e = tid + r*256;
      int k2 = e >> 6, o = e & 63;
      int og = O0 + o;
      v2f val = {};
      if (og < Cout){
        const float* wp = Wt + (size_t)og*Cin2 + c0 + 2*k2;
        val.x = wp[0]; val.y = wp[1];
      }
      sW2[k2][o] = val;
    }
    {                                            // A tile: masked edge features (gather)
      int s = tid >> 1;
      int p = s >> 5;
      float m = sMask[s];
      int ks = s & 31;
      int cb = (tid & 1) * 16;
      const float* ctr = xin + (ptbase + p)*Cin;
      const float* nbr = xin + ((size_t)b*NN + sIdx[p][ks])*Cin;
      #pragma unroll
      for (int cc = 0; cc < 16; ++cc){
        int c = c0 + cb + cc;
        float v = (c < Cin) ? (nbr[c] - ctr[c]) : ctr[c - Cin];
        sF[s][cb+cc] = v * m;
      }
    }
    __syncthreads();
    int row  = lane & 15;
    int koff = (lane >> 4) * 2;
    #pragma unroll
    for (int kc = 0; kc < KC; kc += 4){
      v2f a; a.x = sF[w*16+row][kc+koff]; a.y = sF[w*16+row][kc+koff+1];
      int k2 = (kc + koff) >> 1;                 // even -> single b64 per fragment
      #pragma unroll
      for (int j = 0; j < 4; ++j){
        v2f bb = sW2[k2][j*16+row];
        acc[j] = wmma4(a, bb, acc[j]);
      }
    }
  }

  int col = lane & 15;
  int rbase = (lane < 16) ? 0 : 8;
  if (PHASE == 0){
    #pragma unroll
    for (int j = 0; j < 4; ++j){
      float s1 = 0.f, s2 = 0.f;
      #pragma unroll
      for (int r = 0; r < 8; ++r){ float v = acc[j][r]; s1 += v; s2 += v*v; }
      atomicAdd(&sSum[j*16+col], s1); atomicAdd(&sSq[j*16+col], s2);
    }
    __syncthreads();
    if (tid < 64 && O0 + tid < Cout){
      atomicAdd(&statSum[O0+tid], sSum[tid]);
      atomicAdd(&statSq [O0+tid], sSq [tid]);
    }
  } else {
    #pragma unroll
    for (int j = 0; j < 4; ++j){
      int c = O0 + j*16 + col;
      if (c >= Cout) continue;
      float sc = scale[c], sh = shift[c];
      #pragma unroll
      for (int r = 0; r < 8; ++r){
        int srow = w*16 + r + rbase;
        float v = lrelu(acc[j][r]*sc + sh);
        unsigned key = (sMask[srow] > 0.f) ? fenc(v) : ENC_NEGINF;
        atomicMax(&sMaxKey[srow >> 5][j*16+col], key);
      }
    }
    __syncthreads();
    {
      int p = tid >> 6, c = tid & 63;
      if (O0 + c < Cout){
        size_t pt = ptbase + p;
        float v = okBuf[pt] ? fdec(sMaxKey[p][c]) : 0.f;
        feat_out[pt*Cout + O0 + c] = v;
      }
    }
  }
}

// -------------------------------- point-wise 1x1 conv: WMMA GEMM, h -> ws
// M-tile 128 points, N-tile 64 channels; optional bias; optional masked stats.

template<int DO_STATS>
__global__ __launch_bounds__(256)
void point_gemm(const float* __restrict__ A, const float* __restrict__ Wt, // [Cout][Cin]
                const float* __restrict__ bias, int hasBias,
                float* __restrict__ Hout,
                float* __restrict__ statSum, float* __restrict__ statSq,
                const int* __restrict__ okBuf, int Cin, int Cout){
  __shared__ float sA[128][KC+4];
  __shared__ v2f   sW2[KC/2][64];
  __shared__ float sSum[64], sSq[64];
  __shared__ float sOk[128];
  int tid = threadIdx.x, lane = tid & 31, w = tid >> 5;
  int m0 = blockIdx.x * 128;
  int O0 = blockIdx.y * 64;
  if (tid < 128) sOk[tid] = okBuf[m0+tid] ? 1.f : 0.f;
  if (DO_STATS && tid < 64){ sSum[tid] = 0.f; sSq[tid] = 0.f; }
  __syncthreads();

  v8f acc[4] = {};
  for (int c0 = 0; c0 < Cin; c0 += KC){
    __syncthreads();
    #pragma unroll
    for (int r = 0; r < 4; ++r){
      int e = tid + r*256;
      int k2 = e >> 6, o = e & 63;
      int og = O0 + o;
      v2f val = {};
      if (og < Cout){
        const float* wp = Wt + (size_t)og*Cin + c0 + 2*k2;
        val.x = wp[0]; val.y = wp[1];
      }
      sW2[k2][o] = val;
    }
    {
      int s = tid >> 1, cb = (tid & 1) * 16;
      const float* ap = A + (size_t)(m0+s)*Cin + c0;
      #pragma unroll
      for (int cc = 0; cc < 16; ++cc) sA[s][cb+cc] = ap[cb+cc];
    }
    __syncthreads();
    int row = lane & 15, koff = (lane >> 4) * 2;
    #pragma unroll
    for (int kc = 0; kc < KC; kc += 4){
      v2f a; a.x = sA[w*16+row][kc+koff]; a.y = sA[w*16+row][kc+koff+1];
      int k2 = (kc + koff) >> 1;
      #pragma unroll
      for (int j = 0; j < 4; ++j){
        v2f bb = sW2[k2][j*16+row];
        acc[j] = wmma4(a, bb, acc[j]);
      }
    }
  }

  int col = lane & 15;
  int rbase = (lane < 16) ? 0 : 8;
  #pragma unroll
  for (int j = 0; j < 4; ++j){
    int c = O0 + j*16 + col;
    if (c >= Cout) continue;
    float bv = hasBias ? bias[c] : 0.f;
    float s1 = 0.f, s2 = 0.f;
    #pragma unroll
    for (int r = 0; r < 8; ++r){
      int srow = w*16 + r + rbase;
      float v = acc[j][r] + bv;
      Hout[(size_t)(m0+srow)*Cout + c] = v;
      if (DO_STATS){ float mk = sOk[srow]; s1 += v*mk; s2 += v*v*mk; }
    }
    if (DO_STATS){ atomicAdd(&sSum[j*16+col], s1); atomicAdd(&sSq[j*16+col], s2); }
  }
  if (DO_STATS){
    __syncthreads();
    if (tid < 64 && O0 + tid < Cout){
      atomicAdd(&statSum[O0+tid], sSum[tid]);
      atomicAdd(&statSq [O0+tid], sSq [tid]);
    }
  }
}

// ---------------------------------------------------------------- launcher

extern "C" void kernel_launch(void* const* d_in, const int* in_sizes, int n_in,
                              void* d_out, int out_size, void* d_ws, size_t ws_size,
                              hipStream_t stream){
  (void)in_sizes; (void)n_in; (void)out_size; (void)ws_size;
  const float* x          = (const float*)d_in[0];
  const int*   idx        = (const int*)d_in[1];
  const unsigned char* vd = (const unsigned char*)d_in[2];
  const float* enc_w1 = (const float*)d_in[3];  const float* enc_g1 = (const float*)d_in[4];  const float* enc_b1 = (const float*)d_in[5];
  const float* enc_w2 = (const float*)d_in[6];  const float* enc_g2 = (const float*)d_in[7];  const float* enc_b2 = (const float*)d_in[8];
  const float* enc_w3 = (const float*)d_in[9];  const float* enc_g3 = (const float*)d_in[10]; const float* enc_b3 = (const float*)d_in[11];
  const float* enc_w4 = (const float*)d_in[12]; const float* enc_g4 = (const float*)d_in[13]; const float* enc_b4 = (const float*)d_in[14];
  const float* last_w = (const float*)d_in[15]; const float* last_g = (const float*)d_in[16]; const float* last_b = (const float*)d_in[17];
  const float* emb_w1 = (const float*)d_in[18]; const float* emb_bias1 = (const float*)d_in[19];
  const float* emb_g1 = (const float*)d_in[20]; const float* emb_b1 = (const float*)d_in[21];
  const float* emb_w2 = (const float*)d_in[22]; const float* emb_bias2 = (const float*)d_in[23];
  const float* emb_g2 = (const float*)d_in[24]; const float* emb_b2 = (const float*)d_in[25];
  const float* out_w  = (const float*)d_in[26]; const float* out_bias = (const float*)d_in[27];

  float* ws = (float*)d_ws;
  size_t off = 0;
  auto alloc = [&](size_t n){ float* p = ws + off; off += (n + 63) & ~(size_t)63; return p; };
  float* xt0  = alloc((size_t)BN*3);
  float* xt1  = alloc((size_t)BN*64);
  float* xt2  = alloc((size_t)BN*64);
  float* xt3  = alloc((size_t)BN*128);
  float* xt4  = alloc((size_t)BN*256);
  float* xcat = alloc((size_t)BN*512);
  float* pA   = alloc((size_t)BN*1024);   // h buffer (reused)
  float* pB   = alloc((size_t)BN*1024);   // activation buffer (reused)
  float* pC   = alloc((size_t)BN*512);
  int*   okB  = (int*)alloc(BN);
  float* stSum   = alloc(1024);
  float* stSq    = alloc(1024);
  float* stScale = alloc(1024);
  float* stShift = alloc(1024);
  float* counts  = alloc(8);

  // prologue
  zero_f<<<1, 64, 0, stream>>>(counts, 8);
  transpose_x<<<(BN*3+255)/256, 256, 0, stream>>>(x, xt0);
  ok_kernel<<<(BN+255)/256, 256, 0, stream>>>(vd, okB, counts);

  // ---- edge block 1 (3 -> 64, scalar)
  zero_f<<<(2048+255)/256, 256, 0, stream>>>(stSum, 2048);  // stSum+stSq contiguous
  edge1<0><<<BN/4, 256, 0, stream>>>(xt0, idx, vd, enc_w1, stSum, stSq, nullptr, nullptr, okB, nullptr);
  finalize_bn<<<1, 64, 0, stream>>>(stSum, stSq, counts, 0, enc_g1, enc_b1, stScale, stShift, 64);
  edge1<1><<<BN/4, 256, 0, stream>>>(xt0, idx, vd, enc_w1, nullptr, nullptr, stScale, stShift, okB, xt1);

  // ---- edge blocks 2-4 (WMMA)
  struct { const float* xin; const float* W; const float* g; const float* b; float* xout; int Cin; int Cout; }
  eb[3] = {
    { xt1, enc_w2, enc_g2, enc_b2, xt2,  64,  64 },
    { xt2, enc_w3, enc_g3, enc_b3, xt3,  64, 128 },
    { xt3, enc_w4, enc_g4, enc_b4, xt4, 128, 256 },
  };
  for (int i = 0; i < 3; ++i){
    dim3 g(BN/4, eb[i].Cout/64);
    zero_f<<<(2048+255)/256, 256, 0, stream>>>(stSum, 2048);
    edge_gemm<0><<<g, 256, 0, stream>>>(eb[i].xin, idx, vd, eb[i].W, stSum, stSq,
                                        nullptr, nullptr, okB, nullptr, eb[i].Cin, eb[i].Cout);
    finalize_bn<<<(eb[i].Cout+63)/64, 64, 0, stream>>>(stSum, stSq, counts, 0, eb[i].g, eb[i].b,
                                                       stScale, stShift, eb[i].Cout);
    edge_gemm<1><<<g, 256, 0, stream>>>(eb[i].xin, idx, vd, eb[i].W, nullptr, nullptr,
                                        stScale, stShift, okB, eb[i].xout, eb[i].Cin, eb[i].Cout);
  }

  // ---- concat(64+64+128+256 = 512)
  concat_kernel<<<(BN*512+255)/256, 256, 0, stream>>>(xt1, xt2, xt3, xt4, xcat);

  // ---- last: 512 -> 1024, masked BN + lrelu
  zero_f<<<(2048+255)/256, 256, 0, stream>>>(stSum, 2048);
  point_gemm<1><<<dim3(BN/128, 1024/64), 256, 0, stream>>>(xcat, last_w, nullptr, 0, pA,
                                                           stSum, stSq, okB, 512, 1024);
  finalize_bn<<<(1024+63)/64, 64, 0, stream>>>(stSum, stSq, counts, 1, last_g, last_b, stScale, stShift, 1024);
  bn_act<<<(BN*1024+255)/256, 256, 0, stream>>>(pA, stScale, stShift, pB, BN*1024, 1024);

  // ---- emb1: 1024 -> 512 (+bias), masked BN + lrelu
  zero_f<<<(2048+255)/256, 256, 0, stream>>>(stSum, 2048);
  point_gemm<1><<<dim3(BN/128, 512/64), 256, 0, stream>>>(pB, emb_w1, emb_bias1, 1, pA,
                                                          stSum, stSq, okB, 1024, 512);
  finalize_bn<<<(512+63)/64, 64, 0, stream>>>(stSum, stSq, counts, 1, emb_g1, emb_b1, stScale, stShift, 512);
  bn_act<<<(BN*512+255)/256, 256, 0, stream>>>(pA, stScale, stShift, pC, BN*512, 512);

  // ---- emb2: 512 -> 128 (+bias), masked BN + lrelu
  zero_f<<<(2048+255)/256, 256, 0, stream>>>(stSum, 2048);
  point_gemm<1><<<dim3(BN/128, 128/64), 256, 0, stream>>>(pC, emb_w2, emb_bias2, 1, pA,
                                                          stSum, stSq, okB, 512, 128);
  finalize_bn<<<(128+63)/64, 64, 0, stream>>>(stSum, stSq, counts, 1, emb_g2, emb_b2, stScale, stShift, 128);
  bn_act<<<(BN*128+255)/256, 256, 0, stream>>>(pA, stScale, stShift, pB, BN*128, 128);

  // ---- out: 128 -> 32 (+bias), no BN
  point_gemm<0><<<dim3(BN/128, 1), 256, 0, stream>>>(pB, out_w, out_bias, 1, pA,
                                                     nullptr, nullptr, okB, 128, 32);
  write_out<<<(BN*32+255)/256, 256, 0, stream>>>(pA, (float*)d_out);
}